// VectorQuantizer_5368709120468
// MI455X (gfx1250) — compile-verified
//
#include <hip/hip_runtime.h>

typedef float v2f __attribute__((ext_vector_type(2)));
typedef float v8f __attribute__((ext_vector_type(8)));

#define NUM_EMB   1024
#define DIM       64
#define N_ROWS    131072            // 8*128*128
#define MT        4                 // 16-row WMMA tiles per wave (64 rows/wave)
#define WAVES_PB  4                 // waves per block -> 256 rows per block
#define N_ELEMS   8388608           // N_ROWS * DIM
#define GATHER_BLOCKS 32768         // N_ELEMS / 256

// ---------------------------------------------------------------------------
// Kernel 0: codebook squared norms  ||e_k||^2  -> ws
// ---------------------------------------------------------------------------
__global__ __launch_bounds__(256)
void vq_cb_norms_kernel(const float* __restrict__ cb, float* __restrict__ norms) {
  int r = blockIdx.x * blockDim.x + threadIdx.x;
  if (r < NUM_EMB) {
    const float4* p = (const float4*)(cb + (size_t)r * DIM);
    float s = 0.0f;
#pragma unroll
    for (int i = 0; i < 16; ++i) {
      float4 v = p[i];
      s += v.x * v.x + v.y * v.y + v.z * v.z + v.w * v.w;
    }
    norms[r] = s;
  }
}

// ---------------------------------------------------------------------------
// Kernel 1: WMMA distance matmul + on-the-fly argmin.
// Each wave owns 64 input rows (4 x 16x16 C tiles) with the A panel fully
// register-resident, and sweeps all 1024 codebook columns in 16-wide tiles.
// Distance compared is  ||e||^2 - 2*x.e  (row-constant ||x||^2 dropped).
// ---------------------------------------------------------------------------
__global__ __launch_bounds__(128)
void vq_argmin_kernel(const float* __restrict__ x, const float* __restrict__ cb,
                      const float* __restrict__ norms, int* __restrict__ idx_out,
                      float* __restrict__ fidx_out) {
  const int lane  = threadIdx.x & 31;
  const int wave  = threadIdx.x >> 5;
  const int r0    = blockIdx.x * (MT * 16 * WAVES_PB) + wave * (MT * 16);
  const int mrow  = lane & 15;          // M (or N) index within a 16-wide tile
  const int khalf = (lane >> 4) * 2;    // lanes 0-15: K=4t,4t+1 ; 16-31: 4t+2,4t+3

  // A fragments: a[mt][t] covers rows r0+mt*16+[0..15], depth 4t..4t+3
  v2f a[MT][16];
#pragma unroll
  for (int mt = 0; mt < MT; ++mt) {
    const float* ap = x + (size_t)(r0 + mt * 16 + mrow) * DIM + khalf;
#pragma unroll
    for (int t = 0; t < 16; ++t)
      a[mt][t] = *(const v2f*)(ap + 4 * t);
  }

  float minv[MT][8];
  int   mini[MT][8];
#pragma unroll
  for (int mt = 0; mt < MT; ++mt)
#pragma unroll
    for (int j = 0; j < 8; ++j) { minv[mt][j] = 3.4e38f; mini[mt][j] = 0; }

  for (int c = 0; c < NUM_EMB / 16; ++c) {
    const int col = c * 16 + mrow;                 // this lane's codebook column
    const float* bp = cb + (size_t)col * DIM + khalf;

    v8f acc[MT] = {};
#pragma unroll
    for (int t = 0; t < 16; ++t) {
      v2f b = *(const v2f*)(bp + 4 * t);           // B fragment, reused by 4 WMMAs
#pragma unroll
      for (int mt = 0; mt < MT; ++mt)
        acc[mt] = __builtin_amdgcn_wmma_f32_16x16x4_f32(
            /*neg_a=*/false, a[mt][t], /*neg_b=*/false, b,
            /*c_mod=*/(short)0, acc[mt], /*reuse_a=*/false, /*reuse_b=*/false);
    }

    const float ck = norms[col];
#pragma unroll
    for (int mt = 0; mt < MT; ++mt)
#pragma unroll
      for (int j = 0; j < 8; ++j) {
        float d = ck - 2.0f * acc[mt][j];
        // strict '<' keeps the earliest (lowest) column on ties, like argmin
        if (d < minv[mt][j]) { minv[mt][j] = d; mini[mt][j] = col; }
      }
  }

  // Reduce argmin across the 16 lanes of each half-wave (C-tile N dimension).
#pragma unroll
  for (int mt = 0; mt < MT; ++mt)
#pragma unroll
    for (int j = 0; j < 8; ++j) {
      float v = minv[mt][j];
      int   i = mini[mt][j];
#pragma unroll
      for (int off = 8; off >= 1; off >>= 1) {
        float v2 = __shfl_xor(v, off, 32);
        int   i2 = __shfl_xor(i, off, 32);
        if (v2 < v || (v2 == v && i2 < i)) { v = v2; i = i2; }
      }
      if ((lane & 15) == 0) {
        int row = r0 + mt * 16 + j + ((lane >> 4) ? 8 : 0);  // M = j or j+8
        idx_out[row]  = i;
        fidx_out[row] = (float)i;   // encoding_indices output (as float)
      }
    }
}

// ---------------------------------------------------------------------------
// Kernel 2: gather quantized output + per-block deterministic loss partials
// ---------------------------------------------------------------------------
__global__ __launch_bounds__(256)
void vq_gather_loss_kernel(const float* __restrict__ x, const float* __restrict__ cb,
                           const int* __restrict__ idx, float* __restrict__ out_q,
                           float* __restrict__ partials) {
  __shared__ float sm[256];
  const size_t e = (size_t)blockIdx.x * 256 + threadIdx.x;
  const int row = (int)(e >> 6);
  const int k   = (int)(e & 63);
  const float q  = cb[(size_t)idx[row] * DIM + k];
  const float xv = x[e];
  out_q[e] = q;                       // STE forward value == quantized
  const float d = q - xv;
  sm[threadIdx.x] = d * d;
  __syncthreads();
#pragma unroll
  for (int s = 128; s > 0; s >>= 1) {
    if (threadIdx.x < s) sm[threadIdx.x] += sm[threadIdx.x + s];
    __syncthreads();
  }
  if (threadIdx.x == 0) partials[blockIdx.x] = sm[0];
}

// ---------------------------------------------------------------------------
// Kernel 3: fixed-order final reduction -> vq_loss = 1.25 * MSE
// (q_latent_loss == e_latent_loss value-wise; 1 + COMMITMENT_COST = 1.25)
// ---------------------------------------------------------------------------
__global__ __launch_bounds__(256)
void vq_loss_finish_kernel(const float* __restrict__ partials,
                           float* __restrict__ loss_out) {
  __shared__ float sm[256];
  float s = 0.0f;
  for (int i = threadIdx.x; i < GATHER_BLOCKS; i += 256) s += partials[i];
  sm[threadIdx.x] = s;
  __syncthreads();
#pragma unroll
  for (int st = 128; st > 0; st >>= 1) {
    if (threadIdx.x < st) sm[threadIdx.x] += sm[threadIdx.x + st];
    __syncthreads();
  }
  if (threadIdx.x == 0) *loss_out = 1.25f * sm[0] / (float)N_ELEMS;
}

// ---------------------------------------------------------------------------
extern "C" void kernel_launch(void* const* d_in, const int* in_sizes, int n_in,
                              void* d_out, int out_size, void* d_ws, size_t ws_size,
                              hipStream_t stream) {
  const float* x  = (const float*)d_in[0];   // [8,128,128,64] f32
  const float* cb = (const float*)d_in[1];   // [1024,64] f32

  float* out      = (float*)d_out;
  float* out_q    = out;                     // N_ELEMS floats (quantized)
  float* out_loss = out + N_ELEMS;           // 1 float (vq_loss)
  float* out_idx  = out + N_ELEMS + 1;       // N_ROWS (indices, as float)

  float* norms    = (float*)d_ws;            // 1024 f32
  int*   idxbuf   = (int*)(norms + NUM_EMB); // N_ROWS i32
  float* partials = (float*)(idxbuf + N_ROWS); // GATHER_BLOCKS f32

  vq_cb_norms_kernel<<<NUM_EMB / 256, 256, 0, stream>>>(cb, norms);
  vq_argmin_kernel<<<N_ROWS / (MT * 16 * WAVES_PB), 32 * WAVES_PB, 0, stream>>>(
      x, cb, norms, idxbuf, out_idx);
  vq_gather_loss_kernel<<<GATHER_BLOCKS, 256, 0, stream>>>(x, cb, idxbuf, out_q, partials);
  vq_loss_finish_kernel<<<1, 256, 0, stream>>>(partials, out_loss);
}